// CasPer_77077483094710
// MI455X (gfx1250) — compile-verified
//
#include <hip/hip_runtime.h>

#define N_IN   2048
#define N_HID  17
#define N_OUT  8
#define BATCH  16384
#define NC     32                    // 17 hidden + 8 out + 7 pad columns
#define KT     (N_IN / 4)            // 512 k-tiles of K=4
#define WH_STRIDE (N_IN + N_HID - 1) // 2064
#define WO_STRIDE (N_IN + N_HID)     // 2065

typedef __attribute__((ext_vector_type(2))) float v2f;
typedef __attribute__((ext_vector_type(8))) float v8f;

// ---------------------------------------------------------------------------
// Phase 0: pack combined weight matrix Wc (2048 x 32) into the WMMA B-operand
// lane layout, ktile-major:  Bpack[ktile][ntile][lane][v]  (v = 2 floats/lane)
//   B tile is 4x16 (KxN), f32:  lane 0-15  -> N=lane,   VGPR0=K0, VGPR1=K1
//                               lane 16-31 -> N=lane-16,VGPR0=K2, VGPR1=K3
// Column j<17  -> W_h[j, :2048];  17<=j<25 -> W_out[j-17, :2048];  else 0.
// ---------------------------------------------------------------------------
__global__ void pack_B(const float* __restrict__ W_h,
                       const float* __restrict__ W_out,
                       float* __restrict__ Bpack) {
    int idx = blockIdx.x * blockDim.x + threadIdx.x;   // [0, 512*2*64)
    if (idx >= KT * 2 * 64) return;
    int t    = idx >> 7;        // ktile
    int r    = idx & 127;
    int nt   = r >> 6;          // ntile (N 0-15 / 16-31)
    int lv   = r & 63;
    int lane = lv >> 1;
    int v    = lv & 1;
    int n = nt * 16 + (lane & 15);
    int k = 4 * t + ((lane >> 4) << 1) + v;
    float val = 0.0f;
    if (n < N_HID)                val = W_h[n * WH_STRIDE + k];
    else if (n < N_HID + N_OUT)   val = W_out[(n - N_HID) * WO_STRIDE + k];
    Bpack[idx] = val;
}

// ---------------------------------------------------------------------------
// Phase 1: P = X @ Wc  via V_WMMA_F32_16X16X4_F32.
// Each wave: M=64 rows (4 A-tiles) x N=32 (2 B-tiles) -> 8 v8f accumulators.
// 4 waves per block, 64 blocks -> 256 waves cover all 16384 rows exactly,
// so EXEC is all-ones (required for WMMA).
// ---------------------------------------------------------------------------
__global__ __launch_bounds__(128) void gemm_phase(const float* __restrict__ X,
                                                  const float* __restrict__ Bpack,
                                                  float* __restrict__ P) {
    const int lane     = threadIdx.x & 31;
    const int wave     = blockIdx.x * (blockDim.x >> 5) + (threadIdx.x >> 5);
    const int row_base = wave * 64;

    v8f acc[4][2];
#pragma unroll
    for (int mt = 0; mt < 4; ++mt)
#pragma unroll
        for (int nt = 0; nt < 2; ++nt)
            acc[mt][nt] = (v8f){0.f,0.f,0.f,0.f,0.f,0.f,0.f,0.f};

    const int lrow  = lane & 15;          // A row within tile / B column
    const int khalf = (lane >> 4) << 1;   // 0 or 2: which K pair this lane holds
    const float* a0 = X + (size_t)(row_base + lrow) * N_IN + khalf;
    const v2f*   bp = (const v2f*)Bpack + lane;   // + (t*2+nt)*64 per tile

    for (int t = 0; t < KT; ++t) {
        v2f b0 = bp[(t * 2 + 0) * 64];
        v2f b1 = bp[(t * 2 + 1) * 64];
#pragma unroll
        for (int mt = 0; mt < 4; ++mt) {
            v2f a = *(const v2f*)(a0 + (size_t)mt * 16 * N_IN + t * 4);
            // 8 args: (neg_a, A, neg_b, B, c_mod, C, reuse_a, reuse_b)
            acc[mt][0] = __builtin_amdgcn_wmma_f32_16x16x4_f32(
                false, a, false, b0, (short)0, acc[mt][0], false, false);
            acc[mt][1] = __builtin_amdgcn_wmma_f32_16x16x4_f32(
                false, a, false, b1, (short)0, acc[mt][1], false, false);
        }
    }

    // C/D layout: VGPR r -> row (r + 8*(lane>=16)), col = lane&15 (+16 per ntile)
    const int rtop = (lane >> 4) << 3;
#pragma unroll
    for (int mt = 0; mt < 4; ++mt)
#pragma unroll
        for (int nt = 0; nt < 2; ++nt) {
            int col   = nt * 16 + lrow;
            int rbase = row_base + mt * 16 + rtop;
#pragma unroll
            for (int r = 0; r < 8; ++r)
                P[(size_t)(rbase + r) * NC + col] = acc[mt][nt][r];
        }
}

// ---------------------------------------------------------------------------
// Phase 2: per-row sequential cascade (17 units) + 8-wide head.
// One thread per batch row; cascade weights are wave-uniform (scalar-cached).
// ---------------------------------------------------------------------------
__global__ __launch_bounds__(256) void cascade(const float* __restrict__ P,
                                               const float* __restrict__ W_h,
                                               const float* __restrict__ b_h,
                                               const float* __restrict__ W_out,
                                               const float* __restrict__ b_out,
                                               float* __restrict__ out) {
    int b = blockIdx.x * blockDim.x + threadIdx.x;
    if (b >= BATCH) return;
    const float* Pr = P + (size_t)b * NC;

    float h[N_HID];
#pragma unroll
    for (int j = 0; j < N_HID; ++j) {
        float a = Pr[j] + b_h[j];
#pragma unroll
        for (int i = 0; i < j; ++i)
            a = fmaf(W_h[j * WH_STRIDE + N_IN + i], h[i], a);
        h[j] = 1.0f / (1.0f + __expf(-a));
    }
#pragma unroll
    for (int o = 0; o < N_OUT; ++o) {
        float a = Pr[N_HID + o] + b_out[o];
#pragma unroll
        for (int j = 0; j < N_HID; ++j)
            a = fmaf(W_out[o * WO_STRIDE + N_IN + j], h[j], a);
        out[(size_t)b * N_OUT + o] = a;
    }
}

// ---------------------------------------------------------------------------
extern "C" void kernel_launch(void* const* d_in, const int* in_sizes, int n_in,
                              void* d_out, int out_size, void* d_ws, size_t ws_size,
                              hipStream_t stream) {
    const float* x     = (const float*)d_in[0];
    const float* W_h   = (const float*)d_in[1];
    const float* b_h   = (const float*)d_in[2];
    const float* W_out = (const float*)d_in[3];
    const float* b_out = (const float*)d_in[4];
    float* out = (float*)d_out;

    float* Bpack = (float*)d_ws;                 // 65536 floats (256 KB)
    float* P     = Bpack + KT * 2 * 64;          // 16384*32 floats (2 MB)

    pack_B<<<(KT * 2 * 64 + 255) / 256, 256, 0, stream>>>(W_h, W_out, Bpack);
    gemm_phase<<<BATCH / (64 * 4), 128, 0, stream>>>(x, Bpack, P);
    cascade<<<BATCH / 256, 256, 0, stream>>>(P, W_h, b_h, W_out, b_out, out);
}